// FormantSysth_2001454760579
// MI455X (gfx1250) — compile-verified
//
#include <hip/hip_runtime.h>

// ---------------------------------------------------------------------------
// Fused formant-synthesis forward pass for MI455X (gfx1250, wave32).
// One 256-thread block (8 waves) per 16-column time tile.  The 128x128 MLP
// GEMM runs on V_WMMA_F32_16X16X4_F32 (fp32 matches the reference dtype; the
// problem is latency-bound, not compute-bound, so fp32 WMMA is the right
// precision with zero accuracy cost).
// ---------------------------------------------------------------------------

typedef float v2f __attribute__((ext_vector_type(2)));
typedef float v8f __attribute__((ext_vector_type(8)));

#define BATCH  8
#define TLEN   2048
#define NMELS  64
#define KHARM  30
#define NFORM  6

// mel_scale(hz)  = (log2(hz/440) + 58/24) * 24*64/158
// inv_mel(mel)   = 440 * 2^(mel*158/24 - 58/24)
#define MEL_C1   9.72151898734177f   // 24*64/158
#define MEL_C2   2.41666666666667f   // 58/24
#define LOG2_440 8.78135971352466f
#define IMEL_A   6.58333333333333f   // 158/24

__device__ __forceinline__ float leaky02(float x) { return x > 0.0f ? x : 0.2f * x; }
__device__ __forceinline__ float sgnf(float x)    { return (x > 0.0f) ? 1.0f : ((x < 0.0f) ? -1.0f : 0.0f); }
__device__ __forceinline__ float mel_scale(float hz) {
    return (__log2f(hz) - LOG2_440 + MEL_C2) * MEL_C1;
}
__device__ __forceinline__ float inv_mel(float mel) {
    return 440.0f * exp2f(mel * IMEL_A - MEL_C2);
}

__global__ __launch_bounds__(256)
void formant_synth_kernel(const float* __restrict__ amps,   // (B,2,T)
                          const float* __restrict__ amps_h, // (B,2,T)
                          const float* __restrict__ loud,   // (B,1,T)
                          const float* __restrict__ f0,     // (B,1,T)
                          const float* __restrict__ fh,     // (B,6,T)
                          const float* __restrict__ bh,
                          const float* __restrict__ ah,
                          const float* __restrict__ fn,
                          const float* __restrict__ bn,
                          const float* __restrict__ an,
                          const float* __restrict__ w1,     // (128,1)
                          const float* __restrict__ b1,     // (128)
                          const float* __restrict__ w2,     // (128,128)
                          const float* __restrict__ b2,     // (128)
                          const float* __restrict__ w3,     // (2,128)
                          const float* __restrict__ b3,     // (2)
                          float* __restrict__ out)          // (B,1,T,64)
{
    __shared__ float s_x   [16];        // f0 in Hz per column
    __shared__ float s_h1  [128 * 16];  // h1 tile  [neuron*16 + col]
    __shared__ float s_h2  [128 * 16];  // h2 tile  [neuron*16 + col]
    __shared__ float s_tp  [2 * 16];    // sigmoid outputs
    __shared__ float s_scal[5 * 16];    // amp0, amp1, ah0, ah1, L
    __shared__ float s_form[36 * 16];   // fh[6], bh[6], ah[6], fn[6], bn[6], an[6]

    const int tid = threadIdx.x;
    const int g   = blockIdx.x;         // time-tile index (0..1023)

    // ---------------- Phase 0: stage per-column parameters -----------------
    {
        const int n     = tid & 15;
        const int tf    = g * 16 + n;        // flattened (b,t)
        const int b     = tf >> 11;          // /2048
        const int t     = tf & (TLEN - 1);
        const int base2 = b * 2 * TLEN + t;      // (B,2,T)
        const int base1 = b * TLEN + t;          // (B,1,T)
        const int base6 = b * NFORM * TLEN + t;  // (B,6,T)
        for (int item = tid >> 4; item < 42; item += 16) {
            float v;
            if      (item == 0) v = amps[base2];
            else if (item == 1) v = amps[base2 + TLEN];
            else if (item == 2) v = amps_h[base2];
            else if (item == 3) v = amps_h[base2 + TLEN];
            else if (item == 4) v = loud[base1];
            else if (item == 5) v = inv_mel(f0[base1]);   // f0 in Hz
            else {
                const int j   = item - 6;     // 0..35
                const int grp = j / 6;
                const int jj  = j - grp * 6;
                const float* p = (grp == 0) ? fh : (grp == 1) ? bh : (grp == 2) ? ah
                               : (grp == 3) ? fn : (grp == 4) ? bn : an;
                v = p[base6 + jj * TLEN];
            }
            if      (item < 5)  s_scal[item * 16 + n] = v;
            else if (item == 5) s_x[n] = v;
            else                s_form[(item - 6) * 16 + n] = v;
        }
    }
    __syncthreads();

    // ---------------- Phase 1: h1 = leaky(w1*x + b1) into LDS --------------
    for (int idx = tid; idx < 128 * 16; idx += 256) {
        const int i = idx >> 4;
        const int n = idx & 15;
        s_h1[idx] = leaky02(w1[i] * s_x[n] + b1[i]);
    }
    __syncthreads();

    // ---------------- Phase 2: h2 = leaky(W2 @ h1 + b2) via fp32 WMMA ------
    // Wave w owns output rows [w*16, w*16+16).  16x16x4 f32 WMMA, 32 K-steps.
    {
        const int wave = tid >> 5;
        const int lane = tid & 31;
        const int half = lane >> 4;   // 0: lanes 0-15, 1: lanes 16-31
        const int col  = lane & 15;   // M-lane for A, N for B/C/D
        const int m0   = wave * 16;

        v8f acc = {0.0f, 0.0f, 0.0f, 0.0f, 0.0f, 0.0f, 0.0f, 0.0f};
        const float* w2row = w2 + (m0 + col) * 128;

        #pragma unroll 4
        for (int s = 0; s < 32; ++s) {
            const int ka = s * 4 + half * 2;   // K base for this lane half
            v2f a, bfr;
            a.x   = w2row[ka];                 // A: 16x4, lanes 0-15 K={0,1}, 16-31 K={2,3}
            a.y   = w2row[ka + 1];
            bfr.x = s_h1[ka * 16 + col];       // B: 4x16, mirrored K striping
            bfr.y = s_h1[(ka + 1) * 16 + col];
            acc = __builtin_amdgcn_wmma_f32_16x16x4_f32(
                false, a, false, bfr, (short)0, acc, false, false);
        }
        #pragma unroll
        for (int v = 0; v < 8; ++v) {
            const int row = m0 + v + half * 8; // C/D: VGPR v -> M=v (lo half) / v+8 (hi)
            s_h2[row * 16 + col] = leaky02(acc[v] + b2[row]);
        }
    }
    __syncthreads();

    // ---------------- Phase 3: tp = sigmoid(w3 @ h2 + b3) ------------------
    if (tid < 32) {                 // exactly wave 0, no intra-wave divergence
        const int c = tid >> 4;
        const int n = tid & 15;
        float sum = b3[c];
        const float* wr = w3 + c * 128;
        #pragma unroll 8
        for (int i = 0; i < 128; ++i) sum += wr[i] * s_h2[i * 16 + n];
        s_tp[c * 16 + n] = 1.0f / (1.0f + __expf(-sum));
    }
    __syncthreads();

    // ---------------- Phase 4: harmonic comb + formant masks + mix ---------
    #pragma unroll
    for (int r = 0; r < 4; ++r) {
        const int linear = r * 256 + tid;      // 0..1023 = 16 cols x 64 mels
        const int n = linear >> 6;
        const int f = linear & 63;
        const float ff = (float)f;
        const float x  = s_x[n];               // f0 in Hz

        // hsum over K=30 harmonics (mel constants cancel in the bw difference)
        float hsum = 0.0f;
        for (int k = 1; k <= KHARM; ++k) {
            const float harm = x * (float)k;
            const float bwhz = 24.7f * (harm * 0.00437f + 1.0f);
            const float hi   = harm + 0.5f * bwhz;
            const float lo   = fmaxf(harm - 0.5f * bwhz, 1.0f);
            const float bw   = MEL_C1 * (__log2f(hi) - __log2f(lo));
            const float f0m  = mel_scale(harm);
            const float d    = ff - f0m;
            const float q    = fabsf(d) / (2.5f * bw) - 0.5f;
            const float box  = -sgnf(q) * 0.5f + 0.5f;
            const float u    = d / (1.25f * bw);
            hsum += (1.0f - u * u) * box;
        }

        // envelope from the MLP outputs
        const float av    = s_tp[n];
        const float bv    = s_tp[16 + n];
        const float diff  = ff - av * (float)NMELS;
        const float cond  = sgnf(diff) * 0.5f + 0.5f;
        const float hamon = hsum * (1.0f + (__expf(-0.01f * bv * diff * cond) - 1.0f) * cond);

        // formant Gaussian masks on the Hz grid
        const float ghz = inv_mel(ff * (1.0f / (float)NMELS));
        float mh = 0.0f, mn = 0.0f;
        #pragma unroll
        for (int j = 0; j < NFORM; ++j) {
            const float dh  = ghz - s_form[(0 * 6 + j) * 16 + n];
            const float bwj = s_form[(1 * 6 + j) * 16 + n] + 0.01f;
            mh += s_form[(2 * 6 + j) * 16 + n] * __expf(-0.693f * dh * dh / (2.0f * bwj * bwj));
            const float dn  = ghz - s_form[(3 * 6 + j) * 16 + n];
            const float bwn = s_form[(4 * 6 + j) * 16 + n] + 0.01f;
            mn += s_form[(5 * 6 + j) * 16 + n] * __expf(-0.693f * dn * dn / (2.0f * bwn * bwn));
        }

        const float L    = s_scal[4 * 16 + n];
        const float exch = L * s_scal[0 * 16 + n] * (s_scal[2 * 16 + n] * hamon + s_scal[3 * 16 + n]);
        const float excn = L * s_scal[1 * 16 + n];
        out[(g * 16 + n) * NMELS + f] = exch * mh + excn * mn - 1.0f;
    }
}

extern "C" void kernel_launch(void* const* d_in, const int* in_sizes, int n_in,
                              void* d_out, int out_size, void* d_ws, size_t ws_size,
                              hipStream_t stream) {
    (void)in_sizes; (void)n_in; (void)out_size; (void)d_ws; (void)ws_size;
    formant_synth_kernel<<<(BATCH * TLEN) / 16, 256, 0, stream>>>(
        (const float*)d_in[0],  (const float*)d_in[1],  (const float*)d_in[2],
        (const float*)d_in[3],  (const float*)d_in[4],  (const float*)d_in[5],
        (const float*)d_in[6],  (const float*)d_in[7],  (const float*)d_in[8],
        (const float*)d_in[9],  (const float*)d_in[10], (const float*)d_in[11],
        (const float*)d_in[12], (const float*)d_in[13], (const float*)d_in[14],
        (const float*)d_in[15], (float*)d_out);
}